// Model_1494648619162
// MI455X (gfx1250) — compile-verified
//
#include <hip/hip_runtime.h>
#include <hip/hip_bf16.h>
#include <stdint.h>

#define N_RES 22
#define MAX_RIGID 8
#define MAX_TORSION 7
#define MAX_ATOM 24
#define BLOCK 64
#define FR_STRIDE 97          // dwords per-thread frame slab; 97 co-prime with 64 banks
#define EPSF 1e-12f

// staging layout inside sOPR (reused after the atom stage; units: floats)
#define STG_OPRBB 0                      // BLOCK*12 = 768
#define STG_BB    (BLOCK * 12)           // 768..1535
#define STG_SC    (BLOCK * 24)           // 1536..2431  (BLOCK*14 = 896)

typedef __attribute__((ext_vector_type(4))) unsigned int uint32x4;
typedef __attribute__((ext_vector_type(8))) int          int32x8;
typedef __attribute__((ext_vector_type(4))) int          int32x4;

// ---------------------------------------------------------------------------
// CDNA5 Tensor Data Mover descriptors: 1D (single-line tile) transfers of
// `ndwords` dwords between global and LDS. Built per ISA 8.3-8.6; issued once
// per executing wave (TDM ignores EXEC); tracked by TENSORcnt.
// ---------------------------------------------------------------------------
#if __has_include(<hip/amd_detail/amd_gfx1250_TDM.h>)
#define TDM_LOAD(g0, g1, g2, g3) \
  __builtin_amdgcn_tensor_load_to_lds((g0), (g1), (g2), (g3), (int32x8)(0), 0)
#define TDM_STORE(g0, g1, g2, g3) \
  __builtin_amdgcn_tensor_store_from_lds((g0), (g1), (g2), (g3), (int32x8)(0), 0)
#else
#define TDM_LOAD(g0, g1, g2, g3) \
  __builtin_amdgcn_tensor_load_to_lds((g0), (g1), (g2), (g3), 0)
#define TDM_STORE(g0, g1, g2, g3) \
  __builtin_amdgcn_tensor_store_from_lds((g0), (g1), (g2), (g3), 0)
#endif

__device__ __forceinline__ uint32x4 tdm_group0(const void* lds_ptr,
                                               const void* gptr) {
  const unsigned lds = (unsigned)(uintptr_t)lds_ptr;       // LDS byte offset
  const unsigned long long ga = (unsigned long long)(uintptr_t)gptr;
  // count=1 | lds_addr | global_addr[56:0] | type=2 (bits 127:126)
  uint32x4 g0 = { 1u,
                  lds,
                  (unsigned)(ga & 0xFFFFFFFFull),
                  (unsigned)((ga >> 32) & 0x01FFFFFFull) | 0x80000000u };
  return g0;
}

__device__ __forceinline__ int32x8 tdm_group1(unsigned ndwords) {
  // data_size=4B(code 2); tensor_dim0 = tile_dim0 = ndwords (<=65535 here);
  // tensor_dim1=1; tile_dim1/2=0 (unused); tensor_dim0_stride=ndwords.
  int32x8 g1 = { (int)(2u << 16),
                 (int)((ndwords & 0xFFFFu) << 16),
                 (int)(((ndwords >> 16) & 0xFFFFu) | (1u << 16)),
                 (int)(ndwords << 16),
                 0,
                 (int)ndwords,
                 0,
                 0 };
  return g1;
}

__device__ __forceinline__ void tdm_load_dwords(void* lds_dst, const void* gsrc,
                                                unsigned ndwords) {
  int32x4 g2 = { 1, 1, 0, 0 };               // tensor_dim2=1, tensor_dim3=1
  int32x4 g3 = { 0, (int)(1u << 16), 0, 0 }; // tensor_dim4=1
  TDM_LOAD(tdm_group0(lds_dst, gsrc), tdm_group1(ndwords), g2, g3);
}

__device__ __forceinline__ void tdm_store_dwords(const void* lds_src, void* gdst,
                                                 unsigned ndwords) {
  int32x4 g2 = { 1, 1, 0, 0 };
  int32x4 g3 = { 0, (int)(1u << 16), 0, 0 };
  TDM_STORE(tdm_group0(lds_src, gdst), tdm_group1(ndwords), g2, g3);
}

// ---------------------------------------------------------------------------
// CDNA5 async global->LDS staging (ASYNCcnt path), per-lane 16B copies.
// ---------------------------------------------------------------------------
__device__ __forceinline__ void async_stage_b128(void* lds_dst, const void* gsrc,
                                                 int ndwords, int tid) {
  const int nvec = ndwords >> 2;  // tables are multiples of 4 dwords
  for (int i = tid; i < nvec; i += BLOCK) {
    unsigned int lds = (unsigned int)(uintptr_t)lds_dst + (unsigned int)(i * 16);
    unsigned long long g =
        (unsigned long long)(uintptr_t)gsrc + (unsigned long long)(i * 16);
    asm volatile("global_load_async_to_lds_b128 %0, %1, off"
                 :: "v"(lds), "v"(g) : "memory");
  }
}

__global__ __launch_bounds__(BLOCK) void fold_structure_kernel(
    const float* __restrict__ out0,    // (N,14)
    const float* __restrict__ out1,    // (N,3,3)
    const float* __restrict__ pos,     // (N,3)
    const float* __restrict__ rigidT,  // (22,8,4,3)
    const float* __restrict__ rigidG,  // (22,24,3)
    const int*   __restrict__ rtype,   // (N,)
    const int*   __restrict__ tdep,    // (22,8)
    const int*   __restrict__ gdep,    // (22,24)
    const int*   __restrict__ nrec,    // (1,)
    float* __restrict__ out,           // N*110: R | opr_bb | bb | sc
    int N)
{
  __shared__ __align__(16) float sT [N_RES * MAX_RIGID * 12];  //  8448 B
  __shared__ __align__(16) float sG [N_RES * MAX_ATOM * 3];    //  6336 B
  __shared__ __align__(16) int   sTD[N_RES * MAX_RIGID];       //   704 B
  __shared__ __align__(16) int   sGD[N_RES * MAX_ATOM];        //  2112 B
  __shared__ __align__(16) float sOPR[BLOCK * FR_STRIDE];      // 24832 B (reused)
  __shared__ __align__(16) float sR  [BLOCK * 72];             // 18432 B (total 60864)

  const int tid = threadIdx.x;

  // ---- stage float tables via TDM (wave 0 issues; TENSORcnt) ----
  if (tid == 0) {
    tdm_load_dwords(sT, rigidT, N_RES * MAX_RIGID * 12);
    tdm_load_dwords(sG, rigidG, N_RES * MAX_ATOM * 3);
    __builtin_amdgcn_s_wait_tensorcnt(0);
  }
  // ---- stage int dep tables via async-to-LDS (ASYNCcnt, all lanes) ----
  async_stage_b128(sTD, tdep, N_RES * MAX_RIGID, tid);
  async_stage_b128(sGD, gdep, N_RES * MAX_ATOM, tid);
  asm volatile("s_wait_asynccnt 0" ::: "memory");
  __syncthreads();

  const int n = blockIdx.x * BLOCK + tid;

  // values kept live for the deferred (staged) small-region outputs
  float rot[9];
  float tx = 0.f, ty = 0.f, tz = 0.f;
  float cs[MAX_TORSION], sn[MAX_TORSION];
  float f0[12];

  if (n < N) {
    const int   rt     = rtype[n];
    const float tscale = 0.1f / (float)(*nrec);

    // ---------------- bb frame: Gram–Schmidt on output_1 rows ----------------
    const float* o1 = out1 + (size_t)n * 9;
    const float v0x = o1[0], v0y = o1[1], v0z = o1[2];
    const float v1x = o1[3], v1y = o1[4], v1z = o1[5];
    const float v2x = o1[6], v2y = o1[7], v2z = o1[8];

    const float inv0 = 1.0f / sqrtf(v0x*v0x + v0y*v0y + v0z*v0z + EPSF);
    const float e0x = v0x*inv0, e0y = v0y*inv0, e0z = v0z*inv0;
    const float dd  = e0x*v1x + e0y*v1y + e0z*v1z;
    const float u1x = v1x - dd*e0x, u1y = v1y - dd*e0y, u1z = v1z - dd*e0z;
    const float inv1 = 1.0f / sqrtf(u1x*u1x + u1y*u1y + u1z*u1z + EPSF);
    const float e1x = u1x*inv1, e1y = u1y*inv1, e1z = u1z*inv1;
    const float e2x = e0y*e1z - e0z*e1y;
    const float e2y = e0z*e1x - e0x*e1z;
    const float e2z = e0x*e1y - e0y*e1x;

    // rot row i = (e0[i], e1[i], e2[i])   (columns are e0,e1,e2)
    rot[0] = e0x; rot[1] = e1x; rot[2] = e2x;
    rot[3] = e0y; rot[4] = e1y; rot[5] = e2y;
    rot[6] = e0z; rot[7] = e1z; rot[8] = e2z;
    tx = v2x * tscale; ty = v2y * tscale; tz = v2z * tscale;

    // ---------------- sc: normalized torsion (cos, sin) pairs ----------------
    {
      const float* o0 = out0 + (size_t)n * (2 * MAX_TORSION);
      #pragma unroll
      for (int k = 0; k < MAX_TORSION; ++k) {
        const float x = o0[2*k], y = o0[2*k + 1];
        const float inv = 1.0f / sqrtf(x*x + y*y + EPSF);
        cs[k] = x * inv;
        sn[k] = y * inv;
      }
    }

    // ---------------- initial frames: opr[i] = combine(T[rt][i], Y_i) --------
    float* my = sOPR + tid * FR_STRIDE;            // per-thread LDS frame slab
    const float* T = sT + rt * (MAX_RIGID * 12);   // layout: r[3][3] then t[3]

    // frame 0: Y = [rot ; t + pos]
    {
      const float px = pos[3*(size_t)n], py = pos[3*(size_t)n + 1], pz = pos[3*(size_t)n + 2];
      const float ytx = tx + px, yty = ty + py, ytz = tz + pz;
      const float* X = T;
      #pragma unroll
      for (int i = 0; i < 3; ++i) {
        #pragma unroll
        for (int j = 0; j < 3; ++j) {
          f0[i*3 + j] = X[i*3 + 0]*rot[0*3 + j]
                      + X[i*3 + 1]*rot[1*3 + j]
                      + X[i*3 + 2]*rot[2*3 + j];
        }
        f0[9 + i] = X[i*3 + 0]*ytx + X[i*3 + 1]*yty + X[i*3 + 2]*ytz + X[9 + i];
      }
      #pragma unroll
      for (int k = 0; k < 12; ++k) my[k] = f0[k];
    }

    // frames 1..7: Y = x-axis rotation by torsion k=i-1, zero translation
    #pragma unroll
    for (int i = 1; i < MAX_RIGID; ++i) {
      const float c = cs[i - 1], s = sn[i - 1];
      const float* X = T + i * 12;
      float* dst = my + i * 12;
      #pragma unroll
      for (int r = 0; r < 3; ++r) {
        dst[r*3 + 0] = X[r*3 + 0];
        dst[r*3 + 1] = X[r*3 + 1]*c + X[r*3 + 2]*s;
        dst[r*3 + 2] = X[r*3 + 2]*c - X[r*3 + 1]*s;
        dst[9 + r]   = X[9 + r];
      }
    }

    // ---------------- dependency chain: opr[i] = combine(opr[dep], opr[i]) ---
    #pragma unroll
    for (int i = 1; i < MAX_RIGID; ++i) {
      const int dep = sTD[rt * MAX_RIGID + i];   // dep < i (already finalized)
      const float* P = my + dep * 12;
      float* C = my + i * 12;
      float p[12], cc[12], z[12];
      #pragma unroll
      for (int k = 0; k < 12; ++k) { p[k] = P[k]; cc[k] = C[k]; }
      #pragma unroll
      for (int r = 0; r < 3; ++r) {
        #pragma unroll
        for (int j = 0; j < 3; ++j) {
          z[r*3 + j] = p[r*3 + 0]*cc[0*3 + j]
                     + p[r*3 + 1]*cc[1*3 + j]
                     + p[r*3 + 2]*cc[2*3 + j];
        }
        z[9 + r] = p[r*3 + 0]*cc[9] + p[r*3 + 1]*cc[10] + p[r*3 + 2]*cc[11] + p[9 + r];
      }
      #pragma unroll
      for (int k = 0; k < 12; ++k) C[k] = z[k];
    }

    // ---------------- atoms: R = rot(opr[gdep[a]]) @ G[a] + t  -> LDS stage --
    {
      float* oR = sR + tid * 72;
      const float* Gr = sG + rt * (MAX_ATOM * 3);
      const int*   gd = sGD + rt * MAX_ATOM;
      #pragma unroll 4
      for (int a = 0; a < MAX_ATOM; ++a) {
        const int g = gd[a];
        const float* M = my + g * 12;
        const float gx = Gr[a*3], gy = Gr[a*3 + 1], gz = Gr[a*3 + 2];
        oR[a*3]     = M[0]*gx + M[1]*gy + M[2]*gz + M[9];
        oR[a*3 + 1] = M[3]*gx + M[4]*gy + M[5]*gz + M[10];
        oR[a*3 + 2] = M[6]*gx + M[7]*gy + M[8]*gz + M[11];
      }
    }
  }

  // ---- all slab reads done; reuse sOPR[0..2432) as small-region staging ----
  __syncthreads();
  if (n < N) {
    float* so = sOPR + STG_OPRBB + tid * 12;   // opr_bb = frame 0 (f0)
    float* sb = sOPR + STG_BB    + tid * 12;   // bb = [rot ; t]
    float* ss = sOPR + STG_SC    + tid * (2 * MAX_TORSION);
    #pragma unroll
    for (int k = 0; k < 12; ++k) so[k] = f0[k];
    #pragma unroll
    for (int k = 0; k < 9; ++k) sb[k] = rot[k];
    sb[9] = tx; sb[10] = ty; sb[11] = tz;
    #pragma unroll
    for (int k = 0; k < MAX_TORSION; ++k) {
      ss[2*k]     = cs[k];
      ss[2*k + 1] = sn[k];
    }
  }
  __syncthreads();

  // ---------------- TDM stores: entire output leaves via the DMA engine -----
  if (tid == 0) {
    const int n0 = blockIdx.x * BLOCK;
    int nvalid = N - n0;
    if (nvalid > BLOCK) nvalid = BLOCK;
    if (nvalid > 0) {
      tdm_store_dwords(sR,                out + (size_t)n0 * 72,
                       (unsigned)(nvalid * 72));
      tdm_store_dwords(sOPR + STG_OPRBB,  out + (size_t)N * 72 + (size_t)n0 * 12,
                       (unsigned)(nvalid * 12));
      tdm_store_dwords(sOPR + STG_BB,     out + (size_t)N * 84 + (size_t)n0 * 12,
                       (unsigned)(nvalid * 12));
      tdm_store_dwords(sOPR + STG_SC,     out + (size_t)N * 96 + (size_t)n0 * (2 * MAX_TORSION),
                       (unsigned)(nvalid * (2 * MAX_TORSION)));
      // keep workgroup (and its LDS) alive until the DMA drains
      __builtin_amdgcn_s_wait_tensorcnt(0);
    }
  }
}

extern "C" void kernel_launch(void* const* d_in, const int* in_sizes, int n_in,
                              void* d_out, int out_size, void* d_ws, size_t ws_size,
                              hipStream_t stream) {
  const float* out0   = (const float*)d_in[0];
  const float* out1   = (const float*)d_in[1];
  const float* pos    = (const float*)d_in[2];
  const float* rigidT = (const float*)d_in[3];
  const float* rigidG = (const float*)d_in[4];
  const int*   rtype  = (const int*)  d_in[5];
  const int*   tdep   = (const int*)  d_in[6];
  const int*   gdep   = (const int*)  d_in[7];
  const int*   nrec   = (const int*)  d_in[8];
  float* out = (float*)d_out;

  const int N = in_sizes[5];  // residue_type element count == N_NODES
  const int grid = (N + BLOCK - 1) / BLOCK;
  fold_structure_kernel<<<grid, BLOCK, 0, stream>>>(
      out0, out1, pos, rigidT, rigidG, rtype, tdep, gdep, nrec, out, N);
}